// LSTMWithBN_11871289606585
// MI455X (gfx1250) — compile-verified
//
#include <hip/hip_runtime.h>
#include <hip/hip_bf16.h>

// ---------------------------------------------------------------------------
// BatchNorm-LSTM on MI455X (gfx1250, wave32).
//  Phase 1: Xpre[B*S,1024] = x @ Wx + b      (parallel, fp32 WMMA 16x16x4)
//  Phase 2: persistent kernel, 16 WGs, one global barrier per timestep.
//           Each WG owns 16 hidden features == 64 gate columns, so the
//           per-gate BatchNorm over the 128-row batch is WG-local.
//  Wh preload uses GLOBAL_LOAD_ASYNC_TO_LDS_B32 (ASYNCcnt path).
// ---------------------------------------------------------------------------

#define B_   128
#define S_   1024
#define IN_  128
#define H_   256
#define G4_  1024        // 4*H
#define EPS_ 1e-5f

#define NWG      16      // workgroups in recurrent phase
#define FPW      16      // features per WG
#define COLS     64      // 4 gates * FPW gemm columns per WG
#define THREADS2 512     // 16 wave32 waves

typedef float v2f __attribute__((ext_vector_type(2)));
typedef float v8f __attribute__((ext_vector_type(8)));

// ---------------- Phase 0: zero h, c, barrier counter ----------------------
__global__ void init_state(float* __restrict__ h, float* __restrict__ c,
                           unsigned* __restrict__ counter) {
    int i = blockIdx.x * blockDim.x + threadIdx.x;
    if (i == 0) *counter = 0u;
    const int n = B_ * H_;
    for (int j = i; j < n; j += gridDim.x * blockDim.x) {
        h[j] = 0.f;
        c[j] = 0.f;
    }
}

// ---------------- Phase 1: Xpre = x @ Wx + bias ----------------------------
// One 16x16 output tile per wave, K = 128 (32 fp32 WMMAs per tile).
__global__ void input_gemm(const float* __restrict__ x,     // [B*S, IN]
                           const float* __restrict__ W,     // [IN+H, 4H]
                           const float* __restrict__ bias,  // [4H]
                           float* __restrict__ Xpre) {      // [B*S, 4H]
    const int lane = threadIdx.x & 31;
    const int wave = threadIdx.x >> 5;
    const int tile = blockIdx.x * 4 + wave;   // 4 waves per block
    const int rowTile = tile >> 6;            // 64 column tiles
    const int colTile = tile & 63;
    const int row0 = rowTile * 16;
    const int col0 = colTile * 16;
    const int m    = lane & 15;
    const int koff = (lane >> 4) << 1;        // 0 for lanes 0-15, 2 for 16-31

    v8f acc;
    const float bv = bias[col0 + m];
#pragma unroll
    for (int v = 0; v < 8; ++v) acc[v] = bv;

    const float* arow = x + (size_t)(row0 + m) * IN_;
#pragma unroll 4
    for (int k = 0; k < IN_; k += 4) {
        v2f a = *(const v2f*)(arow + k + koff);         // A[m][k+koff .. +1]
        v2f bf;
        bf.x = W[(size_t)(k + koff)     * G4_ + col0 + m];  // B[k+koff][n]
        bf.y = W[(size_t)(k + koff + 1) * G4_ + col0 + m];
        acc = __builtin_amdgcn_wmma_f32_16x16x4_f32(
            false, a, false, bf, (short)0, acc, false, false);
    }
    const int rbase = row0 + ((lane >> 4) << 3);
#pragma unroll
    for (int v = 0; v < 8; ++v)
        Xpre[(size_t)(rbase + v) * G4_ + col0 + m] = acc[v];
}

// ---------------- Phase 2: persistent recurrent kernel ---------------------
__launch_bounds__(THREADS2, 1)
__global__ void recurrent_lstm(const float* __restrict__ W,      // [384,1024]
                               const float* __restrict__ gamma,  // [4,256]
                               const float* __restrict__ beta,   // [4,256]
                               const float* __restrict__ Xpre,   // [B*S,1024]
                               float* __restrict__ out,          // outputs|hy|cy
                               float* __restrict__ hbuf,         // [B,H]
                               float* __restrict__ cbuf,         // [B,H]
                               unsigned* __restrict__ counter) {
    __shared__ float Wlds[H_ * COLS];          // 64 KB   [k][g*16+f]
    __shared__ float glds[B_ * COLS];          // 32 KB   [bi][g*16+f]
    __shared__ float red[2 * 8 * COLS];        //  4 KB   partial sums/sumsq
    __shared__ float meanv[COLS];
    __shared__ float istdv[COLS];

    const int tid  = threadIdx.x;
    const int lane = tid & 31;
    const int wave = tid >> 5;
    const int wg   = blockIdx.x;               // 0..15
    const int f0   = wg * FPW;                 // global feature base

    // ---- Async preload of this WG's Wh slice into LDS ---------------------
    // Wlds[k*64 + g*16 + f] = W[(128+k)*1024 + g*256 + f0 + f]
    // GVS mode: mem_addr = SADDR(W) + VADDR(32-bit byte offset); VDST holds
    // the LDS byte address (low 32 bits of the flat shared pointer).
    for (int i = tid; i < H_ * COLS; i += THREADS2) {
        const int k = i >> 6, lc = i & 63, g = lc >> 4, f = lc & 15;
        const unsigned ldsb = (unsigned)(unsigned long long)(&Wlds[i]);
        const unsigned goff =
            (unsigned)(((IN_ + k) * G4_ + g * 256 + f0 + f) * 4);
        asm volatile("global_load_async_to_lds_b32 %0, %1, %2"
                     :
                     : "v"(ldsb), "v"(goff), "s"(W)
                     : "memory");
    }
    asm volatile("s_wait_asynccnt 0x0" ::: "memory");
    __syncthreads();

    const int m    = lane & 15;
    const int koff = (lane >> 4) << 1;
    const int rt   = wave & 7;                 // row tile 0..7 (16 rows each)
    const int gp   = wave >> 3;                // gate pair 0 or 1
    const size_t OUT_HY = (size_t)B_ * S_ * H_;
    const size_t OUT_CY = OUT_HY + (size_t)B_ * H_;

    for (int t = 0; t < S_; ++t) {
        // ---- GEMM: g = Xpre[:,t,:] + h @ Wh (this WG's 64 columns) --------
        // Both of this wave's tiles share the SAME A fragment (same h rows,
        // different gate columns): interleave them -> one A load feeds two
        // independent WMMA accumulate chains (ILP on the serial path).
        {
            const int row0 = rt * 16;
            const int rb   = row0 + ((lane >> 4) << 3);
            const int g0   = gp * 2;
            const int g1   = g0 + 1;
            const int l0   = g0 * 16 + m;            // LDS-local columns
            const int l1   = g1 * 16 + m;
            const int c0   = g0 * 256 + f0 + m;      // global gate columns
            const int c1   = g1 * 256 + f0 + m;

            v8f acc0, acc1;                          // C init = Xpre (has bias)
#pragma unroll
            for (int v = 0; v < 8; ++v) {
                const size_t rowoff = ((size_t)(rb + v) * S_ + t) * G4_;
                acc0[v] = Xpre[rowoff + c0];
                acc1[v] = Xpre[rowoff + c1];
            }

            const float* hrow = hbuf + (size_t)(row0 + m) * H_;
#pragma unroll 8
            for (int k = 0; k < H_; k += 4) {
                v2f a = *(const v2f*)(hrow + k + koff);
                v2f b0, b1;
                b0.x = Wlds[(k + koff)     * COLS + l0];
                b0.y = Wlds[(k + koff + 1) * COLS + l0];
                b1.x = Wlds[(k + koff)     * COLS + l1];
                b1.y = Wlds[(k + koff + 1) * COLS + l1];
                acc0 = __builtin_amdgcn_wmma_f32_16x16x4_f32(
                    false, a, false, b0, (short)0, acc0, false, false);
                acc1 = __builtin_amdgcn_wmma_f32_16x16x4_f32(
                    false, a, false, b1, (short)0, acc1, false, false);
            }
#pragma unroll
            for (int v = 0; v < 8; ++v) {
                glds[(rb + v) * COLS + l0] = acc0[v];
                glds[(rb + v) * COLS + l1] = acc1[v];
            }
        }
        // Prefetch next step's Xpre tile rows for this wave (global_prefetch_b8)
        if (t + 1 < S_) {
            const float* p =
                &Xpre[((size_t)(rt * 16 + m) * S_ + (t + 1)) * G4_ + gp * 512 + f0];
            __builtin_prefetch(p, 0, 3);
        }
        __syncthreads();

        // ---- BatchNorm stats: per column over 128 batch rows --------------
        {
            const int col = tid & 63;
            const int seg = tid >> 6;                 // 8 segments x 16 rows
            float s = 0.f, s2 = 0.f;
#pragma unroll
            for (int r = 0; r < 16; ++r) {
                const float v = glds[(seg * 16 + r) * COLS + col];
                s += v; s2 += v * v;
            }
            red[seg * COLS + col]            = s;
            red[8 * COLS + seg * COLS + col] = s2;
        }
        __syncthreads();
        if (tid < COLS) {
            float s = 0.f, s2 = 0.f;
#pragma unroll
            for (int seg = 0; seg < 8; ++seg) {
                s  += red[seg * COLS + tid];
                s2 += red[8 * COLS + seg * COLS + tid];
            }
            const float mn = s * (1.f / 128.f);
            const float vr = s2 * (1.f / 128.f) - mn * mn;
            meanv[tid] = mn;
            istdv[tid] = rsqrtf(vr + EPS_);
        }
        __syncthreads();

        // ---- gates + state update: 128 x 16 elements, 4 per thread --------
        {
            const int f  = tid & 15;
            const int b0 = (tid >> 4) * 4;
            const int fg = f0 + f;
            const float ga0 = gamma[0 * H_ + fg], be0 = beta[0 * H_ + fg];
            const float ga1 = gamma[1 * H_ + fg], be1 = beta[1 * H_ + fg];
            const float ga2 = gamma[2 * H_ + fg], be2 = beta[2 * H_ + fg];
            const float ga3 = gamma[3 * H_ + fg], be3 = beta[3 * H_ + fg];
            const float mn0 = meanv[f],      is0 = istdv[f];
            const float mn1 = meanv[16 + f], is1 = istdv[16 + f];
            const float mn2 = meanv[32 + f], is2 = istdv[32 + f];
            const float mn3 = meanv[48 + f], is3 = istdv[48 + f];
#pragma unroll
            for (int j = 0; j < 4; ++j) {
                const int bi = b0 + j;
                const float gi = (glds[bi * COLS +      f] - mn0) * is0 * ga0 + be0;
                const float gf = (glds[bi * COLS + 16 + f] - mn1) * is1 * ga1 + be1;
                const float gc = (glds[bi * COLS + 32 + f] - mn2) * is2 * ga2 + be2;
                const float go = (glds[bi * COLS + 48 + f] - mn3) * is3 * ga3 + be3;
                const float ig  = 1.f / (1.f + __expf(-gi));
                const float ffg = 1.f / (1.f + __expf(-gf));
                const float e2c = __expf(2.f * gc);
                const float cg  = 1.f - 2.f / (e2c + 1.f);        // tanh
                const float og  = 1.f / (1.f + __expf(-go));
                const float cprev = cbuf[bi * H_ + fg];
                const float cy = ffg * cprev + ig * cg;
                const float e2y = __expf(2.f * cy);
                const float tcy = 1.f - 2.f / (e2y + 1.f);        // tanh
                const float hy = og * tcy;
                cbuf[bi * H_ + fg] = cy;
                hbuf[bi * H_ + fg] = hy;
                out[((size_t)bi * S_ + t) * H_ + fg] = hy;
                if (t == S_ - 1) {
                    out[OUT_HY + (size_t)bi * H_ + fg] = hy;
                    out[OUT_CY + (size_t)bi * H_ + fg] = cy;
                }
            }
        }

        // ---- grid-wide barrier (16 WGs): h/c must be globally visible -----
        __threadfence();
        __syncthreads();
        if (tid == 0) {
            atomicAdd(counter, 1u);
            const unsigned target = (unsigned)(NWG * (t + 1));
            while (__hip_atomic_load(counter, __ATOMIC_ACQUIRE,
                                     __HIP_MEMORY_SCOPE_AGENT) < target)
                __builtin_amdgcn_s_sleep(1);
        }
        __syncthreads();
        __threadfence();
    }
}

// ---------------------------------------------------------------------------
extern "C" void kernel_launch(void* const* d_in, const int* in_sizes, int n_in,
                              void* d_out, int out_size, void* d_ws, size_t ws_size,
                              hipStream_t stream) {
    const float* x     = (const float*)d_in[0];
    const float* W     = (const float*)d_in[1];
    const float* bias  = (const float*)d_in[2];
    const float* gamma = (const float*)d_in[3];
    const float* beta  = (const float*)d_in[4];
    float* out = (float*)d_out;

    float* Xpre = (float*)d_ws;                       // 512 MB scratch
    const size_t xpre_elems = (size_t)B_ * S_ * G4_;
    float* hbuf = Xpre + xpre_elems;
    float* cbuf = hbuf + (size_t)B_ * H_;
    unsigned* counter = (unsigned*)(cbuf + (size_t)B_ * H_);

    init_state<<<64, 256, 0, stream>>>(hbuf, cbuf, counter);

    const int tiles  = (B_ * S_ / 16) * (G4_ / 16);   // 524288
    input_gemm<<<tiles / 4, 128, 0, stream>>>(x, W, bias, Xpre);

    recurrent_lstm<<<NWG, THREADS2, 0, stream>>>(W, gamma, beta, Xpre, out,
                                                 hbuf, cbuf, counter);
}